// GenerateProposals_59433757442720
// MI455X (gfx1250) — compile-verified
//
#include <hip/hip_runtime.h>
#include <stdint.h>

// ---------------- problem constants ----------------
#define H_        400
#define W_        600
#define A_        15
#define HW_       (H_ * W_)            // 240000
#define N_TOT     (H_ * W_ * A_)       // 3600000
#define PRE_NMS   6000
#define POST_NMS  1000
#define NMS_TH    0.7f
#define BBOX_CLIP_F 4.135166556742356f
#define NWORDS    188                  // ceil(6000/32)
#define WSTRIDE   192                  // padded row stride in words
#define CAND_CAP  8192

// scalar slots in workspace
#define SC_T1    0   // coarse threshold bin
#define SC_G1    1   // count strictly above coarse bin
#define SC_NEED2 2   // remaining needed within coarse bin
#define SC_T2    3   // fine threshold bin
#define SC_G2B   4   // count strictly above fine bin (within coarse)
#define SC_CAND  5   // candidate counter
#define SC_KEPT  6   // kept count after NMS
#define SC_NEED1 7   // constant 6000

// ---------------- workspace layout (bytes) ----------------
#define OFF_HIST1   0u                       // 4096  u32
#define OFF_HIST2   16384u                   // 65536 u32
#define OFF_SCAL    278528u                  // 32    u32
#define OFF_CAND    278656u                  // 8192  u64
#define OFF_X1      344192u                  // 6144  f32
#define OFF_Y1      368768u
#define OFF_X2      393344u
#define OFF_Y2      417920u
#define OFF_AREA    442496u
#define OFF_VALIDW  467072u                  // 192 u32
#define OFF_KEPT    467840u                  // 1024 u32
#define OFF_MASK    472064u                  // 6000*192 u32 = 4.6 MB

#define ZERO_WORDS  ((OFF_SCAL / 4u) + 32u)  // hist1 + hist2 + scalars
#define NEED1_WORD  ((OFF_SCAL / 4u) + SC_NEED1)

// monotone float->uint transform (order-preserving for all floats)
__device__ __forceinline__ unsigned mono_u32(float f) {
  unsigned b = __float_as_uint(f);
  return (b & 0x80000000u) ? ~b : (b | 0x80000000u);
}

// ---------------- async global->LDS staging helpers (CDNA5) ----------------
__device__ __forceinline__ void async_tile_issue(unsigned lds_base, unsigned gbase,
                                                 unsigned tid, unsigned long long sbase) {
  // each wave issues exactly 2 async b128 instructions (EXEC-full), so the
  // per-wave ASYNCcnt bookkeeping below (wait <= 2) is exact.
  const unsigned LIMIT = (unsigned)N_TOT * 4u - 16u;
  unsigned g0 = gbase + tid * 16u;            // bytes
  unsigned g1 = g0 + 4096u;
  if (g0 > LIMIT) g0 = LIMIT;                 // clamp tail (data never consumed)
  if (g1 > LIMIT) g1 = LIMIT;
  unsigned l0 = lds_base + tid * 16u;
  unsigned l1 = l0 + 4096u;
  asm volatile("global_load_async_to_lds_b128 %0, %1, %2"
               :: "v"(l0), "v"(g0), "s"(sbase) : "memory");
  asm volatile("global_load_async_to_lds_b128 %0, %1, %2"
               :: "v"(l1), "v"(g1), "s"(sbase) : "memory");
}

// ---------------- kernel 1 (first in file so disasm snippet shows async ops):
// coarse histogram (u >> 20, 4096 bins), double-buffered async LDS staging.
// The prologue async issue is hoisted ABOVE the LDS-hist clear so the first
// tile streams from HBM while the block zeroes its private histogram. --------
__global__ void hist1_kernel(const float* __restrict__ scores, unsigned* __restrict__ hist) {
  __shared__ float tiles[2][2048];        // 2 x 8KB staging
  __shared__ unsigned lhist[4096];        // 16KB private histogram
  const unsigned tid = threadIdx.x;       // 256 threads

  const int nchunks = (N_TOT + 2047) / 2048;   // 1758
  const unsigned long long sbase = (unsigned long long)(uintptr_t)scores;
  const unsigned ldsA = (unsigned)(uintptr_t)(&tiles[0][0]);
  const unsigned ldsB = (unsigned)(uintptr_t)(&tiles[1][0]);

  // kick off the first tile immediately (hides HBM latency behind LDS clear)
  int c = (int)blockIdx.x;
  int buf = 0;
  if (c < nchunks) async_tile_issue(ldsA, (unsigned)c * 8192u, tid, sbase);

  for (unsigned b = tid; b < 4096u; b += 256u) lhist[b] = 0u;
  __syncthreads();

  for (; c < nchunks; c += (int)gridDim.x) {
    int nc = c + (int)gridDim.x;
    if (nc < nchunks) {
      async_tile_issue(buf ? ldsA : ldsB, (unsigned)nc * 8192u, tid, sbase);
      asm volatile("s_wait_asynccnt 2" ::: "memory");   // current tile done
    } else {
      asm volatile("s_wait_asynccnt 0" ::: "memory");
    }
    __syncthreads();

    const int cb = c * 2048;
    const float* tile = buf ? &tiles[1][0] : &tiles[0][0];
#pragma unroll
    for (int part = 0; part < 2; ++part) {
#pragma unroll
      for (int q = 0; q < 4; ++q) {
        int local = part * 1024 + (int)tid * 4 + q;
        int g = cb + local;
        if (g < N_TOT) {
          unsigned u = mono_u32(tile[local]);
          atomicAdd(&lhist[u >> 20], 1u);
        }
      }
    }
    __syncthreads();          // everyone done reading before buffer is reissued
    buf ^= 1;
  }
  for (unsigned b = tid; b < 4096u; b += 256u) {
    unsigned v = lhist[b];
    if (v) atomicAdd(&hist[b], v);
  }
}

// ---------------- kernel 0: clear hist/counters ----------------
__global__ void zero_kernel(unsigned* __restrict__ p) {
  unsigned i = blockIdx.x * 256u + threadIdx.x;
  if (i < ZERO_WORDS) p[i] = (i == NEED1_WORD) ? (unsigned)PRE_NMS : 0u;
}

// ---------------- generic threshold finder (descending scan, 1 block/256 thr) ----------------
__global__ void find_thr_kernel(const unsigned* __restrict__ hist, int nbins,
                                const unsigned* __restrict__ needp,
                                unsigned* __restrict__ outT,
                                unsigned* __restrict__ outAbove,
                                unsigned* __restrict__ outNext) {
  __shared__ unsigned ssum[256];
  __shared__ unsigned sexcl[256];
  const int c = threadIdx.x;
  const int chunk = nbins >> 8;
  unsigned mysum = 0;
  const int p0 = c * chunk;
  for (int q = 0; q < chunk; ++q) mysum += hist[nbins - 1 - (p0 + q)];
  ssum[c] = mysum;
  __syncthreads();
  if (c == 0) {
    unsigned r = 0;
    for (int t = 0; t < 256; ++t) { sexcl[t] = r; r += ssum[t]; }
  }
  __syncthreads();
  const unsigned need = *needp;
  unsigned running = sexcl[c];
  if (running < need && need <= running + mysum) {
    for (int q = 0; q < chunk; ++q) {
      const int bin = nbins - 1 - (p0 + q);
      const unsigned cnt = hist[bin];
      if (running < need && need <= running + cnt) {
        *outT = (unsigned)bin;
        *outAbove = running;
        if (outNext) *outNext = need - running;
      }
      running += cnt;
    }
  }
}

// ---------------- kernel 3: fine histogram within coarse bin (float4 streams) ----------------
__global__ void hist2_kernel(const float* __restrict__ scores,
                             const unsigned* __restrict__ scal,
                             unsigned* __restrict__ hist2) {
  const unsigned T = scal[SC_T1];
  const float4* __restrict__ s4 = (const float4*)scores;
  const int nv = N_TOT / 4;                       // 900000 (exact)
  int i = (int)(blockIdx.x * blockDim.x + threadIdx.x);
  const int stride = (int)(gridDim.x * blockDim.x);
  for (; i < nv; i += stride) {
    const float4 v = s4[i];
    unsigned u;
    u = mono_u32(v.x); if ((u >> 20) == T) atomicAdd(&hist2[(u >> 4) & 0xFFFFu], 1u);
    u = mono_u32(v.y); if ((u >> 20) == T) atomicAdd(&hist2[(u >> 4) & 0xFFFFu], 1u);
    u = mono_u32(v.z); if ((u >> 20) == T) atomicAdd(&hist2[(u >> 4) & 0xFFFFu], 1u);
    u = mono_u32(v.w); if ((u >> 20) == T) atomicAdd(&hist2[(u >> 4) & 0xFFFFu], 1u);
  }
}

// ---------------- kernel 4: compact candidates >= exact threshold (float4) ----------------
__global__ void compact_kernel(const float* __restrict__ scores,
                               unsigned* __restrict__ scal,
                               unsigned long long* __restrict__ cand) {
  const unsigned thr = (scal[SC_T1] << 16) | scal[SC_T2];   // u>>4 granularity
  const float4* __restrict__ s4 = (const float4*)scores;
  const int nv = N_TOT / 4;
  int i = (int)(blockIdx.x * blockDim.x + threadIdx.x);
  const int stride = (int)(gridDim.x * blockDim.x);
  for (; i < nv; i += stride) {
    const float4 v = s4[i];
    const unsigned base = (unsigned)i * 4u;
    unsigned u;
#define COMPACT_ONE(COMP, OFS)                                                  \
    u = mono_u32(COMP);                                                         \
    if ((u >> 4) >= thr) {                                                      \
      unsigned slot = atomicAdd(&scal[SC_CAND], 1u);                            \
      if (slot < CAND_CAP)                                                      \
        cand[slot] = ((unsigned long long)u << 32) |                            \
                     (unsigned)(0xFFFFFFFFu - (base + (OFS)));                  \
    }
    COMPACT_ONE(v.x, 0u)
    COMPACT_ONE(v.y, 1u)
    COMPACT_ONE(v.z, 2u)
    COMPACT_ONE(v.w, 3u)
#undef COMPACT_ONE
  }
}

// ---------------- kernel 5: single-block bitonic sort (desc) of 8192 u64 keys ----------------
__global__ void __launch_bounds__(512) sort_kernel(unsigned long long* __restrict__ cand,
                                                   const unsigned* __restrict__ scal) {
  __shared__ unsigned long long keys[CAND_CAP];   // 64 KB LDS
  unsigned cnt = scal[SC_CAND];
  if (cnt > CAND_CAP) cnt = CAND_CAP;
  for (int i = (int)threadIdx.x; i < CAND_CAP; i += 512)
    keys[i] = (i < (int)cnt) ? cand[i] : 0ull;
  __syncthreads();
  for (int k = 2; k <= CAND_CAP; k <<= 1) {
    for (int j = k >> 1; j > 0; j >>= 1) {
      for (int t = (int)threadIdx.x; t < CAND_CAP; t += 512) {
        const int ixj = t ^ j;
        if (ixj > t) {
          const bool desc = ((t & k) == 0);
          unsigned long long a = keys[t], b = keys[ixj];
          const bool sw = desc ? (a < b) : (a > b);
          if (sw) { keys[t] = b; keys[ixj] = a; }
        }
      }
      __syncthreads();
    }
  }
  for (int i = (int)threadIdx.x; i < CAND_CAP; i += 512) cand[i] = keys[i];
}

// ---------------- kernel 6: decode top-6000 boxes ----------------
__global__ void decode_kernel(const unsigned long long* __restrict__ cand,
                              const float* __restrict__ anchors,
                              const float* __restrict__ bbox,
                              const int* __restrict__ imh_p,
                              const int* __restrict__ imw_p,
                              const int* __restrict__ scale_p,
                              float* __restrict__ x1a, float* __restrict__ y1a,
                              float* __restrict__ x2a, float* __restrict__ y2a,
                              float* __restrict__ areaa,
                              unsigned* __restrict__ validw) {
  const int j = (int)(blockIdx.x * 256 + threadIdx.x);   // 0..6143
  float ox1 = -2.0e9f, oy1 = -2.0e9f, ox2 = -2.1e9f, oy2 = -2.1e9f, area = 1.0f;
  bool valid = false;
  if (j < PRE_NMS) {
    const unsigned long long key = cand[j];
    const unsigned idx = 0xFFFFFFFFu - (unsigned)(key & 0xFFFFFFFFull);
    const int a = (int)(idx % A_);
    const int s = (int)(idx / A_);
    const int wq = s % W_;
    const int hq = s / W_;
    const float sx = (float)wq * 4.0f, sy = (float)hq * 4.0f;
    const float ax1 = anchors[a * 4 + 0] + sx;
    const float ay1 = anchors[a * 4 + 1] + sy;
    const float ax2 = anchors[a * 4 + 2] + sx;
    const float ay2 = anchors[a * 4 + 3] + sy;
    const float wdt = ax2 - ax1 + 1.0f;
    const float hgt = ay2 - ay1 + 1.0f;
    const float cx = ax1 + 0.5f * wdt;
    const float cy = ay1 + 0.5f * hgt;
    const float* dp = bbox + (size_t)(4 * a) * HW_ + (size_t)s;
    const float dx = dp[0];
    const float dy = dp[HW_];
    const float dw = fminf(dp[2 * (size_t)HW_], BBOX_CLIP_F);
    const float dh = fminf(dp[3 * (size_t)HW_], BBOX_CLIP_F);
    const float pcx = dx * wdt + cx, pcy = dy * hgt + cy;
    const float pw = expf(dw) * wdt, ph = expf(dh) * hgt;
    const float imw = (float)(*imw_p), imh = (float)(*imh_p);
    const float minsz = 0.0f * (float)(*scale_p);   // MIN_SIZE * scaling_factor
    ox1 = fminf(fmaxf(pcx - 0.5f * pw, 0.0f), imw - 1.0f);
    oy1 = fminf(fmaxf(pcy - 0.5f * ph, 0.0f), imh - 1.0f);
    ox2 = fminf(fmaxf(pcx + 0.5f * pw - 1.0f, 0.0f), imw - 1.0f);
    oy2 = fminf(fmaxf(pcy + 0.5f * ph - 1.0f, 0.0f), imh - 1.0f);
    const float ws2 = ox2 - ox1 + 1.0f, hs2 = oy2 - oy1 + 1.0f;
    valid = (ws2 >= minsz) && (hs2 >= minsz) &&
            (ox1 + ws2 * 0.5f < imw) && (oy1 + hs2 * 0.5f < imh);
    area = ws2 * hs2;
  }
  x1a[j] = ox1; y1a[j] = oy1; x2a[j] = ox2; y2a[j] = oy2; areaa[j] = area;
  const unsigned long long b = __ballot(valid);
  if ((threadIdx.x & 31u) == 0u) validw[j >> 5] = (unsigned)b;
}

// ---------------- kernel 7: IoU suppression bitmask (wave32 ballot) ----------------
__global__ void mask_kernel(const float* __restrict__ x1a, const float* __restrict__ y1a,
                            const float* __restrict__ x2a, const float* __restrict__ y2a,
                            const float* __restrict__ areaa,
                            unsigned* __restrict__ mask) {
  const int i = (int)blockIdx.x;                 // row 0..5999
  const int wave = (int)(threadIdx.x >> 5);
  const int lane = (int)(threadIdx.x & 31u);
  const int word = (int)blockIdx.y * 8 + wave;   // 0..191
  const int j = word * 32 + lane;                // 0..6143
  const float xi1 = x1a[i], yi1 = y1a[i], xi2 = x2a[i], yi2 = y2a[i], ai = areaa[i];
  const float xj1 = x1a[j], yj1 = y1a[j], xj2 = x2a[j], yj2 = y2a[j], aj = areaa[j];
  const float xx1 = fmaxf(xi1, xj1), yy1 = fmaxf(yi1, yj1);
  const float xx2 = fminf(xi2, xj2), yy2 = fminf(yi2, yj2);
  const float inter = fmaxf(0.0f, xx2 - xx1 + 1.0f) * fmaxf(0.0f, yy2 - yy1 + 1.0f);
  const float iou = inter / (ai + aj - inter);
  const bool sup = (j > i) && (iou > NMS_TH);
  const unsigned long long b = __ballot(sup);
  if (lane == 0) mask[(size_t)i * WSTRIDE + word] = (unsigned)b;
}

// ---------------- kernel 8: sequential greedy NMS, single wave32 ----------------
#define NMS_CHUNK(RK, BASE)                                                   \
  for (int i = (BASE); i < (BASE) + 1024 && i < PRE_NMS; ++i) {               \
    if (i + 1 < PRE_NMS)                                                      \
      __builtin_prefetch(mask + (size_t)(i + 1) * WSTRIDE + lane, 0, 3);      \
    const unsigned wrd = __shfl(RK, (i >> 5) & 31, 32);                       \
    if (!((wrd >> (i & 31)) & 1u)) {                                          \
      const unsigned* row = mask + (size_t)i * WSTRIDE;                       \
      r0 |= row[lane];       r1 |= row[32 + lane];  r2 |= row[64 + lane];     \
      r3 |= row[96 + lane];  r4 |= row[128 + lane]; r5 |= row[160 + lane];    \
    }                                                                         \
  }

__global__ void __launch_bounds__(32) nms_kernel(const unsigned* __restrict__ validw,
                                                 const unsigned* __restrict__ mask,
                                                 unsigned* __restrict__ kept,
                                                 unsigned* __restrict__ scal) {
  __shared__ unsigned words[WSTRIDE];
  const int lane = (int)threadIdx.x;   // wave32
  // removed state: word w = k*32+lane lives in r_k of lane; init = ~valid
  unsigned r0 = (0 * 32 + lane < NWORDS) ? ~validw[0 * 32 + lane] : 0xFFFFFFFFu;
  unsigned r1 = (1 * 32 + lane < NWORDS) ? ~validw[1 * 32 + lane] : 0xFFFFFFFFu;
  unsigned r2 = (2 * 32 + lane < NWORDS) ? ~validw[2 * 32 + lane] : 0xFFFFFFFFu;
  unsigned r3 = (3 * 32 + lane < NWORDS) ? ~validw[3 * 32 + lane] : 0xFFFFFFFFu;
  unsigned r4 = (4 * 32 + lane < NWORDS) ? ~validw[4 * 32 + lane] : 0xFFFFFFFFu;
  unsigned r5 = (5 * 32 + lane < NWORDS) ? ~validw[5 * 32 + lane] : 0xFFFFFFFFu;

  NMS_CHUNK(r0, 0)
  NMS_CHUNK(r1, 1024)
  NMS_CHUNK(r2, 2048)
  NMS_CHUNK(r3, 3072)
  NMS_CHUNK(r4, 4096)
  NMS_CHUNK(r5, 5120)

  words[0 * 32 + lane] = r0; words[1 * 32 + lane] = r1; words[2 * 32 + lane] = r2;
  words[3 * 32 + lane] = r3; words[4 * 32 + lane] = r4; words[5 * 32 + lane] = r5;
  __syncthreads();

  unsigned running = 0;
  for (int w2 = 0; w2 < NWORDS; ++w2) {
    unsigned keepm = ~words[w2];
    if (w2 == NWORDS - 1) keepm &= 0x0000FFFFu;           // j >= 6000 masked off
    const unsigned bit = (keepm >> lane) & 1u;
    const unsigned rank = running + (unsigned)__popc(keepm & ((1u << lane) - 1u));
    if (bit && rank < POST_NMS) kept[rank] = (unsigned)(w2 * 32 + lane);
    running += (unsigned)__popc(keepm);
  }
  if (lane == 0) scal[SC_KEPT] = running;
}

// ---------------- kernel 9: write outputs ----------------
__global__ void output_kernel(const unsigned* __restrict__ scal,
                              const unsigned* __restrict__ kept,
                              const unsigned long long* __restrict__ cand,
                              const float* __restrict__ x1a, const float* __restrict__ y1a,
                              const float* __restrict__ x2a, const float* __restrict__ y2a,
                              float* __restrict__ out) {
  const int k = (int)threadIdx.x;
  if (k >= POST_NMS) return;
  const unsigned cnt = scal[SC_KEPT];
  float p0 = 0.f, p1 = 0.f, p2 = 0.f, p3 = 0.f, sc = 0.f;
  if ((unsigned)k < cnt) {
    const unsigned i = kept[k];
    p0 = x1a[i]; p1 = y1a[i]; p2 = x2a[i]; p3 = y2a[i];
    const unsigned u = (unsigned)(cand[i] >> 32);
    sc = __uint_as_float(u & 0x7FFFFFFFu);     // undo positive-float mono transform
  }
  out[k * 4 + 0] = p0; out[k * 4 + 1] = p1; out[k * 4 + 2] = p2; out[k * 4 + 3] = p3;
  out[4000 + k] = sc;
}

// ---------------- host launcher ----------------
extern "C" void kernel_launch(void* const* d_in, const int* in_sizes, int n_in,
                              void* d_out, int out_size, void* d_ws, size_t ws_size,
                              hipStream_t stream) {
  (void)in_sizes; (void)n_in; (void)out_size; (void)ws_size;
  const float* scores  = (const float*)d_in[0];
  const float* bbox    = (const float*)d_in[1];
  const float* anchors = (const float*)d_in[2];
  const int*   imh_p   = (const int*)d_in[3];
  const int*   imw_p   = (const int*)d_in[4];
  const int*   scale_p = (const int*)d_in[5];
  float* out = (float*)d_out;

  unsigned char* w8 = (unsigned char*)d_ws;
  unsigned*            hist1  = (unsigned*)(w8 + OFF_HIST1);
  unsigned*            hist2  = (unsigned*)(w8 + OFF_HIST2);
  unsigned*            scal   = (unsigned*)(w8 + OFF_SCAL);
  unsigned long long*  cand   = (unsigned long long*)(w8 + OFF_CAND);
  float* x1a   = (float*)(w8 + OFF_X1);
  float* y1a   = (float*)(w8 + OFF_Y1);
  float* x2a   = (float*)(w8 + OFF_X2);
  float* y2a   = (float*)(w8 + OFF_Y2);
  float* areaa = (float*)(w8 + OFF_AREA);
  unsigned* validw = (unsigned*)(w8 + OFF_VALIDW);
  unsigned* kept   = (unsigned*)(w8 + OFF_KEPT);
  unsigned* mask   = (unsigned*)(w8 + OFF_MASK);

  zero_kernel<<<(ZERO_WORDS + 255u) / 256u, 256, 0, stream>>>((unsigned*)d_ws);
  hist1_kernel<<<256, 256, 0, stream>>>(scores, hist1);
  find_thr_kernel<<<1, 256, 0, stream>>>(hist1, 4096, &scal[SC_NEED1],
                                         &scal[SC_T1], &scal[SC_G1], &scal[SC_NEED2]);
  hist2_kernel<<<512, 256, 0, stream>>>(scores, scal, hist2);
  find_thr_kernel<<<1, 256, 0, stream>>>(hist2, 65536, &scal[SC_NEED2],
                                         &scal[SC_T2], &scal[SC_G2B], (unsigned*)nullptr);
  compact_kernel<<<512, 256, 0, stream>>>(scores, scal, cand);
  sort_kernel<<<1, 512, 0, stream>>>(cand, scal);
  decode_kernel<<<24, 256, 0, stream>>>(cand, anchors, bbox, imh_p, imw_p, scale_p,
                                        x1a, y1a, x2a, y2a, areaa, validw);
  mask_kernel<<<dim3(PRE_NMS, 24, 1), 256, 0, stream>>>(x1a, y1a, x2a, y2a, areaa, mask);
  nms_kernel<<<1, 32, 0, stream>>>(validw, mask, kept, scal);
  output_kernel<<<1, 1024, 0, stream>>>(scal, kept, cand, x1a, y1a, x2a, y2a, out);
}